// GraphConvolution_36661840839012
// MI455X (gfx1250) — compile-verified
//
#include <hip/hip_runtime.h>

typedef __attribute__((ext_vector_type(2))) float v2f;
typedef __attribute__((ext_vector_type(8))) float v8f;

#define OUT_F 128
#define IN_F  128

// ---------------------------------------------------------------------------
// Kernel 1: out[n][f] = bias[f]   (re-initialize output each call)
// ---------------------------------------------------------------------------
__global__ void init_bias_kernel(float* __restrict__ out,
                                 const float* __restrict__ bias,
                                 int total) {
    int i = blockIdx.x * blockDim.x + threadIdx.x;
    if (i < total) out[i] = bias[i & (OUT_F - 1)];
}

// ---------------------------------------------------------------------------
// Kernel 2: feats[N,128] = X[N,128] @ W[128,128] via V_WMMA_F32_16X16X4_F32.
// One wave32 computes one 16x16 output tile; K swept in steps of 4.
//
// A 16x4 f32 frag (ISA 7.12.2): lanes 0-15 -> M=lane, {K=0,K=1} in 2 VGPRs;
//                               lanes 16-31 -> M=lane-16, {K=2,K=3}.
// B 4x16 f32 frag (mirror):     lanes 0-15 -> N=lane, {K=0,K=1};
//                               lanes 16-31 -> N=lane-16, {K=2,K=3}.
// C/D 16x16 f32: VGPR v, lanes 0-15 -> row v, lanes 16-31 -> row v+8, col=lane%16.
// ---------------------------------------------------------------------------
__global__ void gemm_wmma_kernel(const float* __restrict__ X,
                                 const float* __restrict__ W,
                                 float* __restrict__ feats,
                                 int N) {
    const int wave = (blockIdx.x * blockDim.x + threadIdx.x) >> 5;
    const int lane = threadIdx.x & 31;
    const int tiles_n = OUT_F / 16;               // 8
    const int m0 = (wave / tiles_n) * 16;
    const int n0 = (wave % tiles_n) * 16;
    if (m0 >= N) return;

    const int lhalf = lane & 15;
    const int khi   = (lane >> 4) << 1;           // 0 for lanes 0-15, 2 for 16-31

    v8f c = {};
    const float* arow = X + (size_t)(m0 + lhalf) * IN_F + khi;   // 8B aligned
    #pragma unroll
    for (int k = 0; k < IN_F; k += 4) {
        v2f a = *(const v2f*)(arow + k);          // X[m0+lhalf][k+khi .. k+khi+1]
        v2f b;
        b.x = W[(size_t)(k + khi)     * OUT_F + n0 + lhalf];
        b.y = W[(size_t)(k + khi + 1) * OUT_F + n0 + lhalf];
        // 8 args: (neg_a, A, neg_b, B, c_mod, C, reuse_a, reuse_b)
        c = __builtin_amdgcn_wmma_f32_16x16x4_f32(false, a, false, b,
                                                  (short)0, c, false, false);
    }

    const int row = m0 + ((lane >> 4) << 3);      // +8 for upper half-wave
    const int col = n0 + lhalf;
    #pragma unroll
    for (int v = 0; v < 8; ++v)
        feats[(size_t)(row + v) * OUT_F + col] = c[v];
}

// ---------------------------------------------------------------------------
// Kernel 3: per-edge gather * alpha, scatter-add into out[dst].
// One wave per edge; lane handles 4 consecutive floats (float4 gather,
// 4 global_atomic_add_f32 no-return). alp computed redundantly per lane
// from the 101-entry alpha table (L0/L2 resident).
// partner(e) = e+T (e<T), e-T (T<=e<2T), e (self-loop tail).
// ---------------------------------------------------------------------------
__global__ void edge_scatter_kernel(const float* __restrict__ feats,
                                    const float* __restrict__ alpha,
                                    const int*  __restrict__ etype,
                                    const int*  __restrict__ src,
                                    const int*  __restrict__ dst,
                                    float* __restrict__ out,
                                    int E, int T) {
    const int e    = (blockIdx.x * blockDim.x + threadIdx.x) >> 5;
    const int lane = threadIdx.x & 31;
    if (e >= E) return;

    const int partner = (e < T) ? (e + T) : ((e < 2 * T) ? (e - T) : e);
    const float alp = alpha[etype[e]] + alpha[etype[partner]];

    const int s = src[e];
    const int d = dst[e];

    const float4 v = *((const float4*)(feats + (size_t)s * OUT_F) + lane);
    float* o = out + (size_t)d * OUT_F + lane * 4;
    atomicAdd(o + 0, v.x * alp);
    atomicAdd(o + 1, v.y * alp);
    atomicAdd(o + 2, v.z * alp);
    atomicAdd(o + 3, v.w * alp);
}

// ---------------------------------------------------------------------------
extern "C" void kernel_launch(void* const* d_in, const int* in_sizes, int n_in,
                              void* d_out, int out_size, void* d_ws, size_t ws_size,
                              hipStream_t stream) {
    const float* X     = (const float*)d_in[0];   // [N, IN_F]
    const float* W     = (const float*)d_in[1];   // [IN_F, OUT_F]
    const float* bias  = (const float*)d_in[2];   // [OUT_F]
    const float* alpha = (const float*)d_in[3];   // [NUM_REL+1, 1]
    const int*   etype = (const int*)d_in[4];     // [E]
    const int*   src   = (const int*)d_in[5];     // [E]
    const int*   dst   = (const int*)d_in[6];     // [E]
    float*       out   = (float*)d_out;           // [N, OUT_F]
    float*       feats = (float*)d_ws;            // [N, OUT_F] scratch

    const int outf = in_sizes[2];                 // 128
    const int inf  = in_sizes[1] / outf;          // 128
    const int N    = in_sizes[0] / inf;           // 50000
    const int E    = in_sizes[4];                 // 1,650,000
    const int T    = (E - N) / 2;                 // 800,000

    // 1) out = broadcast(bias)
    {
        const int total  = N * outf;
        const int blocks = (total + 255) / 256;
        init_bias_kernel<<<blocks, 256, 0, stream>>>(out, bias, total);
    }

    // 2) feats = X @ W  (WMMA fp32)
    {
        const int tiles   = (N / 16) * (outf / 16);   // 3125 * 8 waves
        const int threads = tiles * 32;
        const int blocks  = (threads + 255) / 256;
        gemm_wmma_kernel<<<blocks, 256, 0, stream>>>(X, W, feats, N);
    }

    // 3) scatter-add messages
    {
        const long long threads = (long long)E * 32;
        const int blocks = (int)((threads + 255) / 256);
        edge_scatter_kernel<<<blocks, 256, 0, stream>>>(feats, alpha, etype,
                                                        src, dst, out, E, T);
    }
}